// VectorQuantizer_36129264894076
// MI455X (gfx1250) — compile-verified
//
#include <hip/hip_runtime.h>

typedef float v2f __attribute__((ext_vector_type(2)));
typedef float v8f __attribute__((ext_vector_type(8)));

// Problem constants (from reference)
constexpr int KCODES = 1024;
constexpr int CDIM   = 256;
constexpr int TDIM   = 2048;
constexpr int BDIM   = 32;
constexpr int ROWS_PER_WG = 128;            // rows (t positions) per workgroup
constexpr int NUM_WG = (BDIM * TDIM) / ROWS_PER_WG;  // 512
constexpr int KT     = 128;                 // codes staged in LDS per pass
constexpr int LSTRIDE = CDIM + 2;           // 258 dwords: even (b64 align) + bank spread

constexpr size_t LOSS_POS = (size_t)BDIM * CDIM * TDIM;  // 16777216
constexpr size_t IDX_BASE = LOSS_POS + 1;

// ---------------------------------------------------------------------------
// Kernel 1: codebook squared norms  ee[k] = sum_c codebook[k][c]^2
// ---------------------------------------------------------------------------
__global__ __launch_bounds__(64) void vq_norms(const float* __restrict__ cb,
                                               float* __restrict__ ee) {
    const int k   = blockIdx.x;
    const int tid = threadIdx.x;
    const float4 v = ((const float4*)(cb + (size_t)k * CDIM))[tid];
    float s = v.x * v.x + v.y * v.y + v.z * v.z + v.w * v.w;
    #pragma unroll
    for (int off = 1; off < 32; off <<= 1) s += __shfl_xor(s, off, 32);
    __shared__ float partial[2];
    if ((tid & 31) == 0) partial[tid >> 5] = s;
    __syncthreads();
    if (tid == 0) ee[k] = partial[0] + partial[1];
}

// ---------------------------------------------------------------------------
// Kernel 2: fused distance-GEMM (WMMA f32) + argmin + gather + quantized out
// ---------------------------------------------------------------------------
__global__ __launch_bounds__(256) void vq_main(const float* __restrict__ x,
                                               const float* __restrict__ cb,
                                               const float* __restrict__ ee,
                                               float* __restrict__ out,
                                               float* __restrict__ loss_partial) {
    extern __shared__ float smem[];
    float* sX   = smem;                               // 128 x LSTRIDE  (x tile, row-major)
    float* sCB  = smem + ROWS_PER_WG * LSTRIDE;       // 128 x LSTRIDE  (codebook tile)
    int*   sIdx = (int*)(smem + 2 * ROWS_PER_WG * LSTRIDE);  // 128
    float* sRed = (float*)(sIdx + ROWS_PER_WG);       // 8

    const int tid  = threadIdx.x;
    const int wg   = blockIdx.x;
    const int n0   = wg * ROWS_PER_WG;                // global first row
    const int b    = n0 / TDIM;
    const int t0   = n0 % TDIM;                       // whole WG shares one b
    const float* xbase = x + (size_t)b * CDIM * TDIM + t0;   // x(c, r) = xbase[c*TDIM + r]

    // ---- stage X tile: coalesced global reads, conflict-free LDS writes ----
    {
        const int r = tid & 127;
        const int ch = tid >> 7;                      // 0/1
        for (int i = 0; i < 128; ++i) {
            const int c = i * 2 + ch;
            sX[r * LSTRIDE + c] = xbase[(size_t)c * TDIM + r];
        }
    }
    __syncthreads();

    const int lane = tid & 31;
    const int wave = tid >> 5;
    const int m16  = lane & 15;                       // A row / B,D column within tile
    const int hi   = lane >> 4;                       // k-dim half select
    const int wr0  = wave * 16;                       // wave's first row in tile

    // ---- preload all 64 A fragments (register resident across entire K loop)
    v2f afr[64];
    #pragma unroll
    for (int ct = 0; ct < 64; ++ct) {
        const int c = ct * 4 + hi * 2;
        afr[ct] = *(const v2f*)(&sX[(wr0 + m16) * LSTRIDE + c]);
    }

    float minv[8];
    int   mini[8];
    #pragma unroll
    for (int i = 0; i < 8; ++i) { minv[i] = 3.4e38f; mini[i] = 0; }

    for (int kt = 0; kt < KCODES / KT; ++kt) {        // 8 codebook tiles
        __syncthreads();                              // previous tile fully consumed
        // stage codebook tile [kt*KT, kt*KT+128) x 256, coalesced float4
        {
            const int c4 = (tid & 63) * 4;
            const int kb = tid >> 6;                  // 0..3
            const float* src = cb + (size_t)(kt * KT) * CDIM;
            #pragma unroll 4
            for (int i = 0; i < 32; ++i) {
                const int k = i * 4 + kb;
                const float4 v = *(const float4*)(src + (size_t)k * CDIM + c4);
                float* d = &sCB[k * LSTRIDE + c4];
                d[0] = v.x; d[1] = v.y; d[2] = v.z; d[3] = v.w;
            }
        }
        // prefetch next tile toward caches while we compute
        if (kt + 1 < KCODES / KT)
            __builtin_prefetch(cb + (size_t)(kt + 1) * KT * CDIM + (size_t)tid * 128, 0, 1);
        __syncthreads();

        for (int ks = 0; ks < KT / 16; ++ks) {        // 8 x 16-column sub-tiles
            const int k0 = kt * KT + ks * 16;
            // hoist the column-norm load: its VMEM latency overlaps the WMMAs
            const float eev = ee[k0 + m16];
            const int   kk  = k0 + m16;

            // B(c, k0+j) = codebook[k0+j][c]; lane j<16 -> (c,c+1), j>=16 -> (c+2,c+3)
            const float* brow = &sCB[(ks * 16 + m16) * LSTRIDE + hi * 2];

            // ---- depth-8 software pipeline on the B stream: keep 8
            //      ds_load_b64 in flight so s_wait_dscnt never drains to 0.
            v2f bbuf[8];
            #pragma unroll
            for (int j = 0; j < 8; ++j)
                bbuf[j] = *(const v2f*)(brow + j * 4);

            v8f acc = {0.f, 0.f, 0.f, 0.f, 0.f, 0.f, 0.f, 0.f};
            #pragma unroll
            for (int ct = 0; ct < 64; ++ct) {
                const v2f bcur = bbuf[ct & 7];
                if (ct + 8 < 64)
                    bbuf[ct & 7] = *(const v2f*)(brow + (ct + 8) * 4);
                acc = __builtin_amdgcn_wmma_f32_16x16x4_f32(
                        false, afr[ct], false, bcur, (short)0, acc, false, false);
            }

            #pragma unroll
            for (int i = 0; i < 8; ++i) {
                const float d = fmaf(-2.0f, acc[i], eev);   // dist - ||x||^2 (row const)
                const bool take = d < minv[i];
                minv[i] = take ? d  : minv[i];
                mini[i] = take ? kk : mini[i];
            }
        }
    }

    // ---- cross-lane argmin within each 16-lane column group (ties -> low k)
    #pragma unroll
    for (int i = 0; i < 8; ++i) {
        float v = minv[i];
        int  ix = mini[i];
        #pragma unroll
        for (int off = 1; off < 16; off <<= 1) {
            const float ov = __shfl_xor(v, off, 32);
            const int   oi = __shfl_xor(ix, off, 32);
            if (ov < v || (ov == v && oi < ix)) { v = ov; ix = oi; }
        }
        if (m16 == 0) sIdx[wr0 + 8 * hi + i] = ix;    // row = wr0 + i + 8*hi
    }
    __syncthreads();

    // ---- indices output (as float, coalesced)
    if (tid < ROWS_PER_WG)
        out[IDX_BASE + (size_t)(n0 + tid)] = (float)sIdx[tid];

    // ---- quantized output [B,C,T] + loss partial
    float lacc = 0.f;
    {
        const int r  = tid & 127;
        const int ch = tid >> 7;
        const float* qrow  = cb + (size_t)sIdx[r] * CDIM;
        float*       obase = out + (size_t)b * CDIM * TDIM + t0;
        for (int i = 0; i < 128; ++i) {
            const int c  = i * 2 + ch;
            const float q  = qrow[c];                 // L2-resident gather
            const float xv = sX[r * LSTRIDE + c];     // reuse staged tile
            obase[(size_t)c * TDIM + r] = q;          // coalesced along t
            const float dd = q - xv;
            lacc = fmaf(dd, dd, lacc);
        }
    }
    #pragma unroll
    for (int off = 1; off < 32; off <<= 1) lacc += __shfl_xor(lacc, off, 32);
    if (lane == 0) sRed[wave] = lacc;
    __syncthreads();
    if (tid == 0) {
        float s = 0.f;
        #pragma unroll
        for (int i = 0; i < 8; ++i) s += sRed[i];
        loss_partial[wg] = s;
    }
}

// ---------------------------------------------------------------------------
// Kernel 3: deterministic final loss reduction
// loss = q_loss + 0.25*e_loss = 1.25 * mean((q-x)^2)
// ---------------------------------------------------------------------------
__global__ __launch_bounds__(256) void vq_loss_final(const float* __restrict__ partials,
                                                     float* __restrict__ out) {
    __shared__ float s[256];
    const int tid = threadIdx.x;
    s[tid] = partials[tid] + partials[tid + 256];
    __syncthreads();
    for (int off = 128; off > 0; off >>= 1) {
        if (tid < off) s[tid] += s[tid + off];
        __syncthreads();
    }
    if (tid == 0)
        out[LOSS_POS] = s[0] * (1.25f / ((float)BDIM * TDIM * CDIM));
}

// ---------------------------------------------------------------------------
extern "C" void kernel_launch(void* const* d_in, const int* in_sizes, int n_in,
                              void* d_out, int out_size, void* d_ws, size_t ws_size,
                              hipStream_t stream) {
    (void)in_sizes; (void)n_in; (void)out_size; (void)ws_size;
    const float* x  = (const float*)d_in[0];   // [32, 256, 2048]
    const float* cb = (const float*)d_in[1];   // [1024, 256]
    float* out = (float*)d_out;
    float* ee       = (float*)d_ws;            // 1024 floats
    float* partials = ee + KCODES;             // 512 floats

    vq_norms<<<KCODES, 64, 0, stream>>>(cb, ee);

    const size_t smem = (size_t)(2 * ROWS_PER_WG * LSTRIDE + ROWS_PER_WG + 8) * sizeof(float);
    vq_main<<<NUM_WG, 256, smem, stream>>>(x, cb, ee, out, partials);

    vq_loss_final<<<1, 256, 0, stream>>>(partials, out);
}